// GATFraudDetector_39676907880717
// MI455X (gfx1250) — compile-verified
//
#include <hip/hip_runtime.h>
#include <hip/hip_bf16.h>
#include <math.h>

#define IN_CH   128
#define HID     64
#define HEADS   8
#define NEG_SLOPE 0.2f
#define SOFT_EPS  1e-16f
#define WN      4      // 16-wide N tiles per wave -> 64 output cols per wave

typedef float v2f __attribute__((ext_vector_type(2)));
typedef float v8f __attribute__((ext_vector_type(8)));
typedef unsigned int u32x4 __attribute__((ext_vector_type(4)));
typedef int i32x4 __attribute__((ext_vector_type(4)));
typedef int i32x8 __attribute__((ext_vector_type(8)));

// ---------------------------------------------------------------------------
// f32 WMMA GEMM: C[M,N] = A[M,K] * B[K,N].  M%16==0, N%(64*blockDim.y)==0, K%4==0.
// Block = blockDim.y waves; all waves share one 16-row A slab staged into LDS
// by the Tensor Data Mover (wave 0 issues the descriptor, TENSORcnt-waited).
// Each wave computes a 16x64 strip: one ds_load_b64 A-fragment feeds 4 WMMAs.
// ---------------------------------------------------------------------------
__global__ void gemm_f32_wmma(const float* __restrict__ A,
                              const float* __restrict__ B,
                              float* __restrict__ C,
                              int M, int K, int N) {
  extern __shared__ float sA[];                 // 16 x K tile, at LDS offset 0
  const int lane   = threadIdx.x;               // 0..31
  const int half   = lane >> 4;
  const int l15    = lane & 15;
  const int tile_m = blockIdx.x;
  const int wcol0  = (blockIdx.y * blockDim.y + threadIdx.y) * (16 * WN);

  // ---- TDM: stage A[tile_m*16 .. +16, 0..K) into LDS (one wave issues) ----
  if (threadIdx.y == 0) {
    const unsigned long long ga =
        (unsigned long long)(const void*)(A + (size_t)tile_m * 16 * K);
    const unsigned uK = (unsigned)K, uM = (unsigned)M;
    u32x4 g0 = {};
    i32x8 g1 = {};
    i32x4 gz4 = {};
    g0[0] = 1u;                                          // count=1, load descr.
    g0[1] = 0u;                                          // lds_addr = 0
    g0[2] = (unsigned)(ga & 0xFFFFFFFFull);              // global_addr[31:0]
    g0[3] = (unsigned)((ga >> 32) & 0x01FFFFFFull) | (2u << 30);  // [56:32]|type=2
    g1[0] = (int)(2u << 16);                             // data_size = 4 bytes
    g1[1] = (int)((uK & 0xFFFFu) << 16);                 // tensor_dim0[15:0]=K
    g1[2] = (int)((uK >> 16) | ((uM & 0xFFFFu) << 16));  // dim0 hi | tensor_dim1 lo
    g1[3] = (int)(((uM >> 16) & 0xFFFFu) | ((uK & 0xFFFFu) << 16)); // dim1 hi | tile_dim0=K
    g1[4] = 16;                                          // tile_dim1=16 rows
    g1[5] = (int)uK;                                     // tensor_dim0_stride=K
    g1[6] = 0; g1[7] = 0;
#if __clang_major__ >= 23
    i32x8 gz8 = {};
    __builtin_amdgcn_tensor_load_to_lds(g0, g1, gz4, gz4, gz8, 0);
#else
    __builtin_amdgcn_tensor_load_to_lds(g0, g1, gz4, gz4, 0);
#endif
    __builtin_amdgcn_s_wait_tensorcnt(0);
  }
  __syncthreads();

  // ---- compute 16x64 per wave ----
  v8f acc[WN] = {};
  const int bcol = wcol0 + l15;
  for (int k0 = 0; k0 < K; k0 += 4) {
    const int ka = k0 + half * 2;
    v2f a = *(const v2f*)&sA[l15 * K + ka];              // ds_load_b64
    const float* __restrict__ Br0 = B + (size_t)ka * N;
    const float* __restrict__ Br1 = B + (size_t)(ka + 1) * N;
#pragma unroll
    for (int j = 0; j < WN; ++j) {
      v2f b;
      b.x = Br0[bcol + 16 * j];
      b.y = Br1[bcol + 16 * j];
      acc[j] = __builtin_amdgcn_wmma_f32_16x16x4_f32(false, a, false, b,
                                                     (short)0, acc[j],
                                                     false, false);
    }
  }
  const int orow = tile_m * 16 + half * 8;
#pragma unroll
  for (int j = 0; j < WN; ++j)
#pragma unroll
    for (int r = 0; r < 8; ++r)
      C[(size_t)(orow + r) * N + bcol + 16 * j] = acc[j][r];
}

// ---------------------------------------------------------------------------
__global__ void fill_k(float* __restrict__ p, float v, int n) {
  int i = blockIdx.x * blockDim.x + threadIdx.x;
  if (i < n) p[i] = v;
}

// es[n,h] = <h[n,h,:], a_src[h,:]>, ed likewise.  tid == n*H + h.
__global__ void node_logits_k(const float* __restrict__ Hf,
                              const float* __restrict__ a_src,
                              const float* __restrict__ a_dst,
                              float* __restrict__ es, float* __restrict__ ed,
                              int NH, int H) {
  int tid = blockIdx.x * blockDim.x + threadIdx.x;
  if (tid >= NH) return;
  int h = tid % H;
  const float* __restrict__ hp = Hf + (size_t)tid * HID;
  const float* __restrict__ as = a_src + (size_t)h * HID;
  const float* __restrict__ ad = a_dst + (size_t)h * HID;
  float s = 0.f, d = 0.f;
#pragma unroll 8
  for (int c = 0; c < HID; ++c) { float v = hp[c]; s += v * as[c]; d += v * ad[c]; }
  es[tid] = s; ed[tid] = d;
}

__device__ __forceinline__ void edge_sd(const int* __restrict__ ei, int E,
                                        int idx, int& s, int& d) {
  if (idx < E) { s = ei[idx]; d = ei[E + idx]; }
  else         { s = d = idx - E; }                      // self loops appended
}

__device__ __forceinline__ float edge_logit(const float* es, const float* ed,
                                            int s, int d, int h, int H) {
  float v = es[s * H + h] + ed[d * H + h];
  return v > 0.f ? v : NEG_SLOPE * v;
}

__global__ void edge_max_k(const int* __restrict__ ei, int E, int total,
                           const float* __restrict__ es, const float* __restrict__ ed,
                           float* __restrict__ m, int H) {
  int tid = blockIdx.x * blockDim.x + threadIdx.x;
  if (tid >= total) return;
  int e = tid / H, h = tid - e * H;
  int s, d; edge_sd(ei, E, e, s, d);
  atomicMax(&m[d * H + h], edge_logit(es, ed, s, d, h, H));
}

__global__ void edge_sum_k(const int* __restrict__ ei, int E, int total,
                           const float* __restrict__ es, const float* __restrict__ ed,
                           const float* __restrict__ m, float* __restrict__ dn, int H) {
  int tid = blockIdx.x * blockDim.x + threadIdx.x;
  if (tid >= total) return;
  int e = tid / H, h = tid - e * H;
  int s, d; edge_sd(ei, E, e, s, d);
  float v = edge_logit(es, ed, s, d, h, H);
  atomicAdd(&dn[d * H + h], __expf(v - m[d * H + h]));
}

// One wave per edge: 32-lane coalesced gather of h[src,h,:] and coalesced
// global_atomic_add_f32 bursts into agg[dst,h,:] (HID=64 = 2 x 32 lanes).
__global__ void edge_agg_wave_k(const int* __restrict__ ei, int E, int Etot,
                                const float* __restrict__ es, const float* __restrict__ ed,
                                const float* __restrict__ m, const float* __restrict__ dn,
                                const float* __restrict__ feat, float* __restrict__ agg,
                                int H) {
  const int wid  = blockIdx.x * (blockDim.x >> 5) + (threadIdx.x >> 5);
  const int lane = threadIdx.x & 31;
  if (wid >= Etot) return;                               // wave-uniform exit
  int s, d; edge_sd(ei, E, wid, s, d);
  for (int h = 0; h < H; ++h) {
    float v = es[s * H + h] + ed[d * H + h];
    v = v > 0.f ? v : NEG_SLOPE * v;
    float alpha = __expf(v - m[d * H + h]) / (dn[d * H + h] + SOFT_EPS);
    const float* __restrict__ fr = feat + (size_t)(s * H + h) * HID;
    float* __restrict__ ar = agg + (size_t)(d * H + h) * HID;
    atomicAdd(&ar[lane],      fr[lane]      * alpha);
    atomicAdd(&ar[lane + 32], fr[lane + 32] * alpha);
  }
}

// in-place: a[i] = elu(a[i] + bias[i % cols])
__global__ void bias_elu_k(float* __restrict__ a, const float* __restrict__ bias,
                           int total, int cols) {
  int i = blockIdx.x * blockDim.x + threadIdx.x;
  if (i >= total) return;
  float v = a[i] + bias[i % cols];
  a[i] = v > 0.f ? v : (__expf(v) - 1.f);
}

// out[n,:] = (agg2[n,:] + b2) @ Wc + bc   (Wc is [64,2])
__global__ void final_k(const float* __restrict__ agg2, const float* __restrict__ b2,
                        const float* __restrict__ Wc, const float* __restrict__ bc,
                        float* __restrict__ out, int N) {
  int n = blockIdx.x * blockDim.x + threadIdx.x;
  if (n >= N) return;
  float a0 = bc[0], a1 = bc[1];
  const float* __restrict__ r = agg2 + (size_t)n * HID;
#pragma unroll 8
  for (int c = 0; c < HID; ++c) {
    float v = r[c] + b2[c];
    a0 += v * Wc[2 * c + 0];
    a1 += v * Wc[2 * c + 1];
  }
  out[2 * n + 0] = a0;
  out[2 * n + 1] = a1;
}

// ---------------------------------------------------------------------------
static inline int cdiv(int a, int b) { return (a + b - 1) / b; }

extern "C" void kernel_launch(void* const* d_in, const int* in_sizes, int n_in,
                              void* d_out, int out_size, void* d_ws, size_t ws_size,
                              hipStream_t stream) {
  const float* x      = (const float*)d_in[0];
  const int*   ei     = (const int*)  d_in[1];   // [2,E] (src row, dst row)
  const float* W1     = (const float*)d_in[2];
  const float* a_src1 = (const float*)d_in[3];
  const float* a_dst1 = (const float*)d_in[4];
  const float* b1     = (const float*)d_in[5];
  const float* W2     = (const float*)d_in[6];
  const float* a_src2 = (const float*)d_in[7];
  const float* a_dst2 = (const float*)d_in[8];
  const float* b2     = (const float*)d_in[9];
  const float* Wc     = (const float*)d_in[10];
  const float* bc     = (const float*)d_in[11];
  float* out = (float*)d_out;

  const int N    = in_sizes[0] / IN_CH;   // 10000
  const int E    = in_sizes[1] / 2;       // 320000
  const int Etot = E + N;                 // with self loops
  const int F1   = HEADS * HID;           // 512

  // workspace layout (floats)
  float* ws = (float*)d_ws;
  size_t o = 0;
  float* h1   = ws + o; o += (size_t)N * F1;     // 10000x512
  float* es1  = ws + o; o += (size_t)N * HEADS;
  float* ed1  = ws + o; o += (size_t)N * HEADS;
  float* m1   = ws + o; o += (size_t)N * HEADS;
  float* dn1  = ws + o; o += (size_t)N * HEADS;
  float* agg1 = ws + o; o += (size_t)N * F1;     // becomes ELU'd layer-2 input
  float* h2   = ws + o; o += (size_t)N * HID;
  float* es2  = ws + o; o += (size_t)N;
  float* ed2  = ws + o; o += (size_t)N;
  float* m2   = ws + o; o += (size_t)N;
  float* dn2  = ws + o; o += (size_t)N;
  float* agg2 = ws + o; o += (size_t)N * HID;

  const int T = 256;

  // ---- init accumulators ----
  fill_k<<<cdiv(N * F1,  T), T, 0, stream>>>(agg1, 0.f,       N * F1);
  fill_k<<<cdiv(N * HEADS,T), T, 0, stream>>>(m1,  -INFINITY, N * HEADS);
  fill_k<<<cdiv(N * HEADS,T), T, 0, stream>>>(dn1, 0.f,       N * HEADS);
  fill_k<<<cdiv(N * HID, T), T, 0, stream>>>(agg2, 0.f,       N * HID);
  fill_k<<<cdiv(N, T),       T, 0, stream>>>(m2,  -INFINITY,  N);
  fill_k<<<cdiv(N, T),       T, 0, stream>>>(dn2, 0.f,        N);

  // ---- layer 1: h1 = x @ W1  (TDM-staged A, WMMA f32) ----
  {
    dim3 blk(32, 4);                                 // 4 waves x 64 cols = 256
    dim3 grd(N / 16, F1 / (16 * WN * 4));            // (625, 2)
    gemm_f32_wmma<<<grd, blk, 16 * IN_CH * sizeof(float), stream>>>(
        x, W1, h1, N, IN_CH, F1);
  }
  node_logits_k<<<cdiv(N * HEADS, T), T, 0, stream>>>(h1, a_src1, a_dst1, es1, ed1,
                                                      N * HEADS, HEADS);
  {
    const int tot = Etot * HEADS;
    edge_max_k<<<cdiv(tot, T), T, 0, stream>>>(ei, E, tot, es1, ed1, m1, HEADS);
    edge_sum_k<<<cdiv(tot, T), T, 0, stream>>>(ei, E, tot, es1, ed1, m1, dn1, HEADS);
    edge_agg_wave_k<<<cdiv(Etot, T / 32), T, 0, stream>>>(ei, E, Etot, es1, ed1,
                                                          m1, dn1, h1, agg1, HEADS);
  }
  bias_elu_k<<<cdiv(N * F1, T), T, 0, stream>>>(agg1, b1, N * F1, F1);

  // ---- layer 2: h2 = elu_out @ W2  (TDM-staged A, WMMA f32) ----
  {
    dim3 blk(32, 1);                                 // 1 wave covers N=64
    dim3 grd(N / 16, 1);
    gemm_f32_wmma<<<grd, blk, 16 * F1 * sizeof(float), stream>>>(
        agg1, W2, h2, N, F1, HID);
  }
  node_logits_k<<<cdiv(N, T), T, 0, stream>>>(h2, a_src2, a_dst2, es2, ed2, N, 1);
  {
    edge_max_k<<<cdiv(Etot, T), T, 0, stream>>>(ei, E, Etot, es2, ed2, m2, 1);
    edge_sum_k<<<cdiv(Etot, T), T, 0, stream>>>(ei, E, Etot, es2, ed2, m2, dn2, 1);
    edge_agg_wave_k<<<cdiv(Etot, T / 32), T, 0, stream>>>(ei, E, Etot, es2, ed2,
                                                          m2, dn2, h2, agg2, 1);
  }

  // ---- classifier ----
  final_k<<<cdiv(N, T), T, 0, stream>>>(agg2, b2, Wc, bc, out, N);
}